// WeightedHausdorffDistance_10952166604856
// MI455X (gfx1250) — compile-verified
//
#include <hip/hip_runtime.h>

typedef __attribute__((ext_vector_type(2))) float v2f;
typedef __attribute__((ext_vector_type(8))) float v8f;

#define HH 256
#define WW 256
#define NPIX (HH * WW)      // 65536
#define PP 512
#define NSLICES 64
#define NPIXTILES (NPIX / 16)  // 4096 tiles of 16 pixels (each within one row)
#define NPTTILES (PP / 16)     // 32 tiles of 16 points

// ws layout (floats):
//   [0]                dmax (global max distance; exact via corner argument)
//   [1..3]             padding (zeroed)
//   [4 .. 4+512)       S[p] = sum over pixels of wd^(-5)
//   [516 .. 516+65536) per-pixel min SQUARED distance, stored as uint bits
//                      (positive floats: uint order == float order)

__global__ void k_init(float* ws) {
    int gid = blockIdx.x * blockDim.x + threadIdx.x;
    unsigned* minb = (unsigned*)(ws + 516);
    if (gid < NPIX) minb[gid] = 0x7F800000u;  // +inf
    if (gid < PP) ws[4 + gid] = 0.0f;
    if (gid == 0) { ws[1] = 0.0f; ws[2] = 0.0f; ws[3] = 0.0f; }
}

// Global max distance: distance to a fixed point is convex over the grid box,
// so the max over the integer grid [0,255]^2 is attained at one of 4 corners.
__global__ void k_dmax(const float* __restrict__ pts, float* ws) {
    __shared__ float red[128];
    int t = threadIdx.x;
    float m = 0.0f;
    for (int p = t; p < PP; p += 128) {
        float py = pts[2 * p + 0];
        float px = pts[2 * p + 1];
#pragma unroll
        for (int c = 0; c < 4; ++c) {
            float cy = (c & 1) ? 255.0f : 0.0f;
            float cx = (c & 2) ? 255.0f : 0.0f;
            float dy = cy - py, dx = cx - px;
            m = fmaxf(m, dy * dy + dx * dx);
        }
    }
    red[t] = m;
    __syncthreads();
    for (int s = 64; s > 0; s >>= 1) {
        if (t < s) red[t] = fmaxf(red[t], red[t + s]);
        __syncthreads();
    }
    if (t == 0) ws[0] = __builtin_amdgcn_sqrtf(red[0]);
}

// Each wave owns one 16-point tile (fixed operand) and strides over pixel
// tiles. Two v_wmma_f32_16x16x4_f32 per tile pair:
//   D  = A(pixels) x B(points): per-lane column = fixed point  -> wd^-5 sums
//   D^T= B x A (operand swap) : per-lane column = pixel        -> min over points
// Symmetric encoding: dist^2 = [y,x,|g|^2,1] . [-2py,-2px,1,|p|^2]; under the
// identical A/B lane packing, swapping operands transposes the tile.
__global__ __launch_bounds__(256) void k_main(const float* __restrict__ hm,
                                              const float* __restrict__ pts,
                                              float* ws) {
    __shared__ float4 feat[PP];
    const int tid = threadIdx.x;

    // Precompute point features once per block (8 KB -> LDS).
    for (int p = tid; p < PP; p += 256) {
        float py = pts[2 * p + 0];
        float px = pts[2 * p + 1];
        feat[p] = make_float4(-2.0f * py, -2.0f * px, py * py + px * px, 0.0f);
    }
    __syncthreads();

    const int lane = tid & 31;
    const int wave = tid >> 5;
    const int gw = blockIdx.x * 8 + wave;
    const int ptTile = gw & (NPTTILES - 1);  // fixed 16-point tile for this wave
    const int slice = gw >> 5;               // 0..NSLICES-1 pixel slice
    const int nlo = lane & 15;
    const bool hi = lane >= 16;

    const float dmax = ws[0];
    float* S = ws + 4;
    unsigned* minb = (unsigned*)(ws + 516);

    // Point operand (K stripes: lanes 0-15 hold K=0,1; lanes 16-31 hold K=2,3).
    float4 f = feat[ptTile * 16 + nlo];
    v2f b;
    b[0] = hi ? 1.0f : f.x;  // K0 = -2*py | K2 = 1
    b[1] = hi ? f.z : f.y;   // K1 = -2*px | K3 = |p|^2

    float acc = 0.0f;  // per-lane sum of wd^-5 for point (ptTile*16 + nlo)

    for (int pt = slice; pt < NPIXTILES; pt += NSLICES) {
        const int pixBase = pt << 4;               // 16 pixels, same grid row
        const float yrow = (float)(pixBase >> 8);  // y constant within the tile
        const float x = (float)((pixBase & 255) + nlo);
        const float gn2 = yrow * yrow + x * x;

        v2f a;
        a[0] = hi ? gn2 : yrow;  // K0 = y | K2 = |g|^2
        a[1] = hi ? 1.0f : x;    // K1 = x | K3 = 1

        v8f c = {};
        // M=pixels, N=points: rows are pixels, lane column is this wave's point.
        v8f d2 = __builtin_amdgcn_wmma_f32_16x16x4_f32(
            false, a, false, b, (short)0, c, false, false);
        // Transposed tile (M=points, N=pixels) via operand swap.
        v8f d2t = __builtin_amdgcn_wmma_f32_16x16x4_f32(
            false, b, false, a, (short)0, c, false, false);

        // ---- term 1: per-pixel min over this wave's 16 points (d^2 space) ----
        // Lane L holds points 0-7 (lo half) or 8-15 (hi half) of the SAME pixel
        // column pixBase+nlo; both halves issue the atomicMin and the atomic
        // unit performs the half-merge. No shuffle, no EXEC divergence.
        float m2 = d2t[0];
#pragma unroll
        for (int r = 1; r < 8; ++r) m2 = fminf(m2, d2t[r]);
        atomicMin(&minb[pixBase + nlo], __float_as_uint(m2));

        // ---- term 2: accumulate wd^-5 over the 8 pixel rows ----
        const int rowBase = pixBase + (hi ? 8 : 0);
        const float4* hm4 = (const float4*)(hm + rowBase);  // 32B aligned
        float4 h0 = hm4[0];
        float4 h1 = hm4[1];
        float hv[8] = {h0.x, h0.y, h0.z, h0.w, h1.x, h1.y, h1.z, h1.w};

#pragma unroll
        for (int r = 0; r < 8; ++r) {
            float d = __builtin_amdgcn_sqrtf(d2[r]);
            // wd = hm*d + (1-hm)*dmax ; accumulate wd^-5 = rcp(wd^5)
            float wd = fmaf(hv[r], d - dmax, dmax);
            float w2 = wd * wd;
            float w4 = w2 * w2;
            acc += __builtin_amdgcn_rcpf(w4 * wd);
        }
    }

    // Both halves accumulated the same point column over disjoint pixel rows.
    acc += __shfl_xor(acc, 16, 32);
    if (lane < 16) atomicAdd(&S[ptTile * 16 + lane], acc);
}

__global__ __launch_bounds__(1024) void k_final(const float* __restrict__ hm,
                                                float* ws,
                                                float* __restrict__ out) {
    __shared__ float red[1024];
    int t = threadIdx.x;
    const unsigned* minb = (const unsigned*)(ws + 516);

    float s1 = 0.0f, s2 = 0.0f;
    for (int i = t; i < NPIX; i += 1024) {
        float h = hm[i];
        float mn = __builtin_amdgcn_sqrtf(__uint_as_float(minb[i]));  // min dist
        s1 = fmaf(h, mn, s1);
        s2 += fabsf(h);
    }
    float s3 = 0.0f;
    if (t < PP) {
        float sm = ws[4 + t] * (1.0f / (float)NPIX);  // mean over pixels
        // soft_min = sm^(1/ALPHA) = sm^(-0.2) = exp2(-0.2 * log2(sm))
        s3 = __builtin_amdgcn_exp2f(-0.2f * __builtin_amdgcn_logf(sm));
    }

    red[t] = s1;
    __syncthreads();
    for (int s = 512; s > 0; s >>= 1) {
        if (t < s) red[t] += red[t + s];
        __syncthreads();
    }
    float sum1 = red[0];
    __syncthreads();

    red[t] = s2;
    __syncthreads();
    for (int s = 512; s > 0; s >>= 1) {
        if (t < s) red[t] += red[t + s];
        __syncthreads();
    }
    float sum2 = red[0];
    __syncthreads();

    red[t] = s3;
    __syncthreads();
    for (int s = 512; s > 0; s >>= 1) {
        if (t < s) red[t] += red[t + s];
        __syncthreads();
    }
    if (t == 0) out[0] = sum1 / sum2 + red[0] * (1.0f / (float)PP);
}

extern "C" void kernel_launch(void* const* d_in, const int* in_sizes, int n_in,
                              void* d_out, int out_size, void* d_ws, size_t ws_size,
                              hipStream_t stream) {
    (void)in_sizes; (void)n_in; (void)out_size; (void)ws_size;
    const float* hm = (const float*)d_in[0];   // (256,256) f32
    const float* pts = (const float*)d_in[1];  // (512,2)  f32
    float* ws = (float*)d_ws;
    float* out = (float*)d_out;

    k_init<<<NPIX / 256, 256, 0, stream>>>(ws);
    k_dmax<<<1, 128, 0, stream>>>(pts, ws);
    k_main<<<(NPTTILES * NSLICES) / 8, 256, 0, stream>>>(hm, pts, ws);
    k_final<<<1, 1024, 0, stream>>>(hm, ws, out);
}